// LightGCN_51994874085596
// MI455X (gfx1250) — compile-verified
//
#include <hip/hip_runtime.h>
#include <hip/hip_bf16.h>

typedef float v8f __attribute__((ext_vector_type(8)));
typedef float v2f __attribute__((ext_vector_type(2)));

#define EMB 64

// Low 32 bits of a generic pointer to LDS == LDS byte offset (flat-address LDS
// mapping); used as VDST for async global->LDS loads.
__device__ __forceinline__ unsigned lds_lo32(const void* p) {
  return (unsigned)(unsigned long long)p;
}

// Hardware f32 atomic add, no-return form (STOREcnt), device scope so the
// RMW resolves in L2 and is coherent across workgroups.
__device__ __forceinline__ void atomic_add_f32_dev(float* p, float v) {
  asm volatile("global_atomic_add_f32 %0, %1, off scope:SCOPE_DEV"
               :: "v"(p), "v"(v) : "memory");
}
__device__ __forceinline__ void atomic_add_f32_dev_off(float* p, float v, int byteoff) {
  // byteoff must be an immediate at each call site after inlining/unrolling;
  // keep it in the address instead (safe for any toolchain):
  asm volatile("global_atomic_add_f32 %0, %1, off scope:SCOPE_DEV"
               :: "v"((float*)((char*)p + byteoff)), "v"(v) : "memory");
}

// -------------------------------------------------------------------------
// init: xA = concat(user,item); acc = same; xB = 0   (float4-vectorized)
// -------------------------------------------------------------------------
__global__ __launch_bounds__(256) void lgcn_init(const float4* __restrict__ user4,
                                                 const float4* __restrict__ item4,
                                                 float4* __restrict__ xA,
                                                 float4* __restrict__ acc,
                                                 float4* __restrict__ xB,
                                                 long user_n4, long total4) {
  long i = (long)blockIdx.x * blockDim.x + threadIdx.x;
  if (i >= total4) return;
  float4 v = (i < user_n4) ? user4[i] : item4[i - user_n4];
  xA[i] = v;
  acc[i] = v;
  xB[i] = make_float4(0.f, 0.f, 0.f, 0.f);
}

// -------------------------------------------------------------------------
// SpMM: y[rows[e]] += vals[e] * x[cols[e]]  (y pre-zeroed)
// Edge chunk (256 edges = 3 KB) streamed to LDS with async b128 copies
// (ASYNCcnt), then 16 lanes/edge: b128 gather from L2-resident x, scale,
// 4 hardware f32 atomics into L2-resident y.
// -------------------------------------------------------------------------
__global__ __launch_bounds__(256) void lgcn_spmm(const int* __restrict__ rows,
                                                 const int* __restrict__ cols,
                                                 const float* __restrict__ vals,
                                                 const float* __restrict__ x,
                                                 float* __restrict__ y,
                                                 long nnz) {
  __shared__ __align__(16) int   s_rows[256];
  __shared__ __align__(16) int   s_cols[256];
  __shared__ __align__(16) float s_vals[256];

  const int t = threadIdx.x;
  const long base = (long)blockIdx.x * 256;

  if (base + 256 <= nnz) {
    // Full chunk: 192 lanes each async-copy 16B (rows | cols | vals).
    if (t < 192) {
      const int which = t >> 6;        // 0=rows, 1=cols, 2=vals
      const int q = (t & 63) * 4;      // element index within chunk
      const void* gp;
      unsigned lp;
      if (which == 0)      { gp = rows + base + q; lp = lds_lo32(&s_rows[q]); }
      else if (which == 1) { gp = cols + base + q; lp = lds_lo32(&s_cols[q]); }
      else                 { gp = vals + base + q; lp = lds_lo32(&s_vals[q]); }
      asm volatile("global_load_async_to_lds_b128 %0, %1, off"
                   :: "v"(lp), "v"(gp) : "memory");
    }
  } else if (base + t < nnz) {
    // Tail chunk (not hit for NNZ % 256 == 0): guarded b32 async copies.
    unsigned lr = lds_lo32(&s_rows[t]);
    unsigned lc = lds_lo32(&s_cols[t]);
    unsigned lv = lds_lo32(&s_vals[t]);
    const int*   pr = rows + base + t;
    const int*   pc = cols + base + t;
    const float* pv = vals + base + t;
    asm volatile("global_load_async_to_lds_b32 %0, %1, off" :: "v"(lr), "v"(pr) : "memory");
    asm volatile("global_load_async_to_lds_b32 %0, %1, off" :: "v"(lc), "v"(pc) : "memory");
    asm volatile("global_load_async_to_lds_b32 %0, %1, off" :: "v"(lv), "v"(pv) : "memory");
  }
  asm volatile("s_wait_asynccnt 0" ::: "memory");
  __syncthreads();

  const int group = t >> 4;  // 16 groups of 16 lanes
  const int sub   = t & 15;  // 4 floats per lane -> full 64-dim row
  long nloc = nnz - base;
  if (nloc > 256) nloc = 256;

  if (nloc == 256) {
#pragma unroll 4
    for (int i = 0; i < 16; ++i) {
      const int e = group * 16 + i;
      const int   r = s_rows[e];
      const int   c = s_cols[e];
      const float v = s_vals[e];
      const float4 xv = *(const float4*)(x + (size_t)c * EMB + sub * 4);
      float* dst = y + (size_t)r * EMB + sub * 4;
      atomic_add_f32_dev_off(dst, v * xv.x, 0);
      atomic_add_f32_dev_off(dst, v * xv.y, 4);
      atomic_add_f32_dev_off(dst, v * xv.z, 8);
      atomic_add_f32_dev_off(dst, v * xv.w, 12);
    }
  } else {
    for (int i = 0; i < 16; ++i) {
      const int e = group * 16 + i;
      if (e >= (int)nloc) break;
      const int   r = s_rows[e];
      const int   c = s_cols[e];
      const float v = s_vals[e];
      const float4 xv = *(const float4*)(x + (size_t)c * EMB + sub * 4);
      float* dst = y + (size_t)r * EMB + sub * 4;
      atomic_add_f32_dev_off(dst, v * xv.x, 0);
      atomic_add_f32_dev_off(dst, v * xv.y, 4);
      atomic_add_f32_dev_off(dst, v * xv.z, 8);
      atomic_add_f32_dev_off(dst, v * xv.w, 12);
    }
  }
}

// -------------------------------------------------------------------------
// combine: acc = (acc + delta) * scale per 16x16 tile on the matrix pipe:
// C <- sum_k A_k*B_k + C with B_k a column selector (exact elementwise add,
// fp32, no rounding difference). Memory-bound, so the WMMA chain is free.
// Also zeroes the buffer used as the next layer's SpMM output.
// -------------------------------------------------------------------------
__global__ __launch_bounds__(256) void lgcn_combine(float* __restrict__ acc,
                                                    const float* __restrict__ delta,
                                                    float* __restrict__ tozero,
                                                    float scale, long total) {
  const int t    = threadIdx.x;
  const int lane = t & 31;
  const int wave = t >> 5;
  const int lo   = lane & 15;   // N for C/D, M for A
  const int hi   = lane >> 4;   // selects M+8 half for C/D, K half for A/B

  const long tile = (long)blockIdx.x * 8 + wave;
  const long base = tile * 256;

  if (base + 256 <= total) {
    // C/D layout: VGPR j, lane -> element (M = j + 8*hi, N = lo)
    v8f c;
#pragma unroll
    for (int j = 0; j < 8; ++j)
      c[j] = acc[base + (long)(j + 8 * hi) * 16 + lo];

#pragma unroll
    for (int k = 0; k < 4; ++k) {
      // A 16x4 layout: VGPR j, lane -> (M = lo, K = j + 2*hi); A_k = X[:,4k:4k+4]
      v2f a, b;
      a[0] = delta[base + (long)lo * 16 + 4 * k + 2 * hi + 0];
      a[1] = delta[base + (long)lo * 16 + 4 * k + 2 * hi + 1];
      // B_k[K][N] = (N == 4k+K); B 4x16 lane layout assumed to mirror A's
      b[0] = (lo == (4 * k + 2 * hi + 0)) ? 1.0f : 0.0f;
      b[1] = (lo == (4 * k + 2 * hi + 1)) ? 1.0f : 0.0f;
      c = __builtin_amdgcn_wmma_f32_16x16x4_f32(false, a, false, b,
                                                (short)0, c, false, false);
    }

#pragma unroll
    for (int j = 0; j < 8; ++j)
      acc[base + (long)(j + 8 * hi) * 16 + lo] = c[j] * scale;
  }

  // zero 8 floats of the next layer's output buffer
  long zbase = ((long)blockIdx.x * 256 + t) * 8;
  if (zbase + 8 <= total) {
    float4 z = make_float4(0.f, 0.f, 0.f, 0.f);
    *(float4*)(tozero + zbase)     = z;
    *(float4*)(tozero + zbase + 4) = z;
  }
}

// -------------------------------------------------------------------------
extern "C" void kernel_launch(void* const* d_in, const int* in_sizes, int n_in,
                              void* d_out, int out_size, void* d_ws, size_t ws_size,
                              hipStream_t stream) {
  const int*   rows     = (const int*)d_in[0];
  const int*   cols     = (const int*)d_in[1];
  const float* vals     = (const float*)d_in[2];
  const float* user_emb = (const float*)d_in[3];
  const float* item_emb = (const float*)d_in[4];

  float* acc = (float*)d_out;            // running sum lives in the output
  const long nnz   = in_sizes[0];        // 12,800,000 edges
  const long userF = in_sizes[3];        // 6.4M floats
  const long itemF = in_sizes[4];        // 6.4M floats
  const long total = userF + itemF;      // 12.8M floats (51.2 MB)

  float* xA = (float*)d_ws;              // ping
  float* xB = xA + total;                // pong (total ws use: 102.4 MB)

  const int initBlocks = (int)((total / 4 + 255) / 256);
  const int spmmBlocks = (int)((nnz + 255) / 256);
  const int combBlocks = (int)((total + 2047) / 2048);   // 8 waves * 256 floats

  lgcn_init<<<initBlocks, 256, 0, stream>>>(
      (const float4*)user_emb, (const float4*)item_emb,
      (float4*)xA, (float4*)acc, (float4*)xB, userF / 4, total / 4);

  // layer 1: xB = A*xA ; acc += xB ; zero xA
  lgcn_spmm<<<spmmBlocks, 256, 0, stream>>>(rows, cols, vals, xA, xB, nnz);
  lgcn_combine<<<combBlocks, 256, 0, stream>>>(acc, xB, xA, 1.0f, total);

  // layer 2: xA = A*xB ; acc += xA ; zero xB
  lgcn_spmm<<<spmmBlocks, 256, 0, stream>>>(rows, cols, vals, xB, xA, nnz);
  lgcn_combine<<<combBlocks, 256, 0, stream>>>(acc, xA, xB, 1.0f, total);

  // layer 3: xB = A*xA ; acc = (acc + xB) * 0.25 ; (zero xA, harmless)
  lgcn_spmm<<<spmmBlocks, 256, 0, stream>>>(rows, cols, vals, xA, xB, nnz);
  lgcn_combine<<<combBlocks, 256, 0, stream>>>(acc, xB, xA, 0.25f, total);
}